// GeometricBridge_14834817040764
// MI455X (gfx1250) — compile-verified
//
#include <hip/hip_runtime.h>

// GeometricBridge pipeline for MI455X (gfx1250, wave32).
// Inputs : x (32,128,64) f32, K (32,128,128) f32, N (32,128,128) f32
// Outputs: z (32,128,64) | v (32,128,64) | g (32,128,128) | Gamma (32,128,128,128)
//          concatenated flat in d_out (float32).

typedef __attribute__((ext_vector_type(2))) float v2f;
typedef __attribute__((ext_vector_type(4))) float v4f;
typedef __attribute__((ext_vector_type(8))) float v8f;
typedef __attribute__((ext_vector_type(4))) unsigned int u32x4;
typedef __attribute__((ext_vector_type(4))) int i32x4;
typedef __attribute__((ext_vector_type(8))) int i32x8;

#define ALPHA 0.1f
#define B_ 32
#define T_ 128
#define D_ 64
#define T2_ 16384   // 128*128
#define TD_ 8192    // 128*64

#ifndef __has_builtin
#define __has_builtin(x) 0
#endif
#if __has_builtin(__builtin_amdgcn_tensor_load_to_lds) && \
    __has_builtin(__builtin_amdgcn_s_wait_tensorcnt)
#define HAVE_TDM_LD 1
#else
#define HAVE_TDM_LD 0
#endif
#if __has_builtin(__builtin_amdgcn_tensor_store_from_lds) && \
    __has_builtin(__builtin_amdgcn_s_wait_tensorcnt)
#define HAVE_TDM_ST 1
#else
#define HAVE_TDM_ST 0
#endif

// ---------------------------------------------------------------------------
// TDM 1-D tile descriptor (ISA 8.3/8.4): contiguous nElems f32 global<->LDS.
//   group0: count=1 | lds_addr | global_addr[56:0] | type=2
//   group1: data_size=4B; tensor_dim0 = tile_dim0 = nElems (<=65535)
// Issued from wave 0 only (TDM ignores EXEC; one instruction per wave).
// ---------------------------------------------------------------------------
#if HAVE_TDM_LD || HAVE_TDM_ST
__device__ __forceinline__ void tdm_desc(u32x4& g0, i32x8& g1,
                                         const void* gptr, const void* lptr,
                                         unsigned nElems) {
  const unsigned long long ga = (unsigned long long)(size_t)gptr;
  const unsigned la = (unsigned)(size_t)lptr;  // flat LDS addr: low 32 bits
  g0[0] = 1u;                                        // count=1, user descriptor
  g0[1] = la;                                        // lds_addr
  g0[2] = (unsigned)ga;                              // global_addr[31:0]
  g0[3] = (unsigned)((ga >> 32) & 0x1FFFFFFull) | (2u << 30);  // [56:32]|type=2
  g1[0] = (int)(2u << 16);                           // data_size = 4 bytes
  g1[1] = (int)((nElems & 0xFFFFu) << 16);           // tensor_dim0[15:0] @63:48
  g1[2] = (int)((nElems >> 16) & 0xFFFFu);           // tensor_dim0[31:16] @79:64
  g1[3] = (int)((nElems & 0xFFFFu) << 16);           // tile_dim0 @127:112
  g1[4] = 0;                                         // tile_dim1=0, tile_dim2=0
  g1[5] = (int)nElems;                               // dim0 stride (benign)
  g1[6] = 0; g1[7] = 0;
}
#endif

__device__ __forceinline__ void g2l_async(const float* g, float* l, int n) {
#if HAVE_TDM_LD
  if ((threadIdx.x >> 5) == 0) {
    u32x4 g0; i32x8 g1; i32x4 zz = {0, 0, 0, 0};
    tdm_desc(g0, g1, g, l, (unsigned)n);
#if __clang_major__ >= 23
    i32x8 z8 = {0, 0, 0, 0, 0, 0, 0, 0};
    __builtin_amdgcn_tensor_load_to_lds(g0, g1, zz, zz, z8, 0);
#else
    __builtin_amdgcn_tensor_load_to_lds(g0, g1, zz, zz, 0);
#endif
  }
#else
  for (int i = threadIdx.x; i < (n >> 2); i += blockDim.x)
    ((v4f*)l)[i] = ((const v4f*)g)[i];
#endif
}

__device__ __forceinline__ void l2g_async(float* g, const float* l, int n) {
#if HAVE_TDM_ST
  if ((threadIdx.x >> 5) == 0) {
    u32x4 g0; i32x8 g1; i32x4 zz = {0, 0, 0, 0};
    tdm_desc(g0, g1, g, l, (unsigned)n);
#if __clang_major__ >= 23
    i32x8 z8 = {0, 0, 0, 0, 0, 0, 0, 0};
    __builtin_amdgcn_tensor_store_from_lds(g0, g1, zz, zz, z8, 0);
#else
    __builtin_amdgcn_tensor_store_from_lds(g0, g1, zz, zz, 0);
#endif
  }
#else
  for (int i = threadIdx.x; i < (n >> 2); i += blockDim.x)
    ((v4f*)g)[i] = ((const v4f*)l)[i];
#endif
}

// Wait for outstanding TDM ops (wave 0), then workgroup barrier.
__device__ __forceinline__ void copy_fence() {
#if HAVE_TDM_LD || HAVE_TDM_ST
  if ((threadIdx.x >> 5) == 0) __builtin_amdgcn_s_wait_tensorcnt(0);
#endif
  __syncthreads();
}
__device__ __forceinline__ void tdm_drain() {
#if HAVE_TDM_LD || HAVE_TDM_ST
  if ((threadIdx.x >> 5) == 0) __builtin_amdgcn_s_wait_tensorcnt(0);
#endif
}

// ---------------------------------------------------------------------------
// 128x128 (x ncols) LDS tile matmul on fp32 WMMA (V_WMMA_F32_16X16X4_F32),
// 256-thread (8-wave) block. Fragment loads are batched in chunks of 16
// K-steps so the ds_load stream is issued ahead of a single s_wait_dscnt,
// then 16 WMMAs run back-to-back on the accumulation chain.
// Fragment layout per ISA 7.12.2:
//   A 16x4 : lanes 0-15 -> M=lane, K={k,k+1}; lanes 16-31 -> M=lane-16, K={k+2,k+3}
//   B 4x16 : mirrored; C 16x16: VGPR r, lanes 0-15 -> M=r, 16-31 -> M=r+8.
// ---------------------------------------------------------------------------
template <bool TA, bool TB>
__device__ __forceinline__ void mm128_wmma(const float* __restrict__ A, int lda,
                                           const float* __restrict__ B, int ldb,
                                           float* __restrict__ C, int ldc,
                                           int ncols) {
  const int wave = threadIdx.x >> 5;
  const int lane = threadIdx.x & 31;
  const int lo   = lane & 15;
  const int koff = (lane >> 4) << 1;  // 0 or 2
  const int ntn  = ncols >> 4;
  const int ntiles = 8 * ntn;
  for (int t = wave; t < ntiles; t += 8) {
    const int mt = (t / ntn) << 4;
    const int nt = (t % ntn) << 4;
    v8f acc = {0.f, 0.f, 0.f, 0.f, 0.f, 0.f, 0.f, 0.f};
#pragma unroll
    for (int kc = 0; kc < 128; kc += 64) {
      v2f af[16], bf[16];
#pragma unroll
      for (int u = 0; u < 16; ++u) {
        const int ka = kc + (u << 2) + koff;
        if (TA) { af[u].x = A[ka * lda + mt + lo];  af[u].y = A[(ka + 1) * lda + mt + lo]; }
        else    { af[u].x = A[(mt + lo) * lda + ka]; af[u].y = A[(mt + lo) * lda + ka + 1]; }
        if (TB) { bf[u].x = B[(nt + lo) * ldb + ka]; bf[u].y = B[(nt + lo) * ldb + ka + 1]; }
        else    { bf[u].x = B[ka * ldb + nt + lo];  bf[u].y = B[(ka + 1) * ldb + nt + lo]; }
      }
#pragma unroll
      for (int u = 0; u < 16; ++u)
        acc = __builtin_amdgcn_wmma_f32_16x16x4_f32(false, af[u], false, bf[u],
                                                    (short)0, acc, false, false);
    }
    const int mrow = mt + ((lane >> 4) << 3);
#pragma unroll
    for (int r = 0; r < 8; ++r)
      C[(mrow + r) * ldc + nt + lo] = acc[r];
  }
}

// ---------------------------------------------------------------------------
// Kernel 1: per-batch Cholesky N = L L^T (lower) and triangular inverse L^{-1}.
// One 128-thread block per batch; both matrices in LDS (128 KB dynamic).
// ---------------------------------------------------------------------------
__global__ void k_chol_inv(const float* __restrict__ Ng, float* __restrict__ LinvG) {
  extern __shared__ float sm[];
  float* Lm = sm;            // 16384 floats
  float* Xm = sm + T2_;      // 16384 floats
  const int b = blockIdx.x, tid = threadIdx.x;
  g2l_async(Ng + (size_t)b * T2_, Lm, T2_);
  copy_fence();

  // Left-looking Cholesky, column by column.
  for (int j = 0; j < 128; ++j) {
    if (tid == 0) {
      float s = Lm[j * 128 + j];
      for (int k = 0; k < j; ++k) { float l = Lm[j * 128 + k]; s -= l * l; }
      Lm[j * 128 + j] = sqrtf(fmaxf(s, 1e-20f));
    }
    __syncthreads();
    const float d = Lm[j * 128 + j];
    for (int i = j + 1 + tid; i < 128; i += 128) {
      float s = Lm[i * 128 + j];
      for (int k = 0; k < j; ++k) s -= Lm[i * 128 + k] * Lm[j * 128 + k];
      Lm[i * 128 + j] = s / d;
    }
    __syncthreads();
  }

  // Column-parallel forward substitution: thread j owns column j of L^{-1};
  // each column depends only on its own earlier rows -> no barriers needed.
  {
    const int j = tid;
    for (int i = 0; i < 128; ++i) {
      if (i >= j) {
        float s = (i == j) ? 1.f : 0.f;
        for (int k = j; k < i; ++k) s -= Lm[i * 128 + k] * Xm[k * 128 + j];
        Xm[i * 128 + j] = s / Lm[i * 128 + i];
      } else {
        Xm[i * 128 + j] = 0.f;
      }
    }
  }
  __syncthreads();
  l2g_async(LinvG + (size_t)b * T2_, Xm, T2_);
  tdm_drain();
}

// ---------------------------------------------------------------------------
// Kernel 2: A = sym( L^{-1} K L^{-T} )   (two WMMA matmuls, 192 KB LDS)
// ---------------------------------------------------------------------------
__global__ void k_form_A(const float* __restrict__ LinvG,
                         const float* __restrict__ Kg,
                         float* __restrict__ Ag) {
  extern __shared__ float sm[];
  float* bl = sm;             // Linv
  float* bk = sm + T2_;       // K, then A
  float* bm = sm + 2 * T2_;   // M1 = Linv*K, then sym(A)
  const int b = blockIdx.x, tid = threadIdx.x;
  g2l_async(LinvG + (size_t)b * T2_, bl, T2_);
  g2l_async(Kg + (size_t)b * T2_, bk, T2_);
  copy_fence();
  mm128_wmma<false, false>(bl, 128, bk, 128, bm, 128, 128);  // M1 = Linv*K
  __syncthreads();
  mm128_wmma<false, true>(bm, 128, bl, 128, bk, 128, 128);   // A = M1*Linv^T
  __syncthreads();
  for (int i = tid; i < T2_; i += 256) {
    const int r = i >> 7, c = i & 127;
    bm[i] = 0.5f * (bk[i] + bk[c * 128 + r]);  // re-symmetrize
  }
  __syncthreads();
  l2g_async(Ag + (size_t)b * T2_, bm, T2_);
  tdm_drain();
}

// ---------------------------------------------------------------------------
// Kernel 3: parallel cyclic Jacobi eigensolver (8 sweeps x 127 steps of 64
// disjoint Givens rotations), then bitonic sort of eigenvalues ascending and
// column-gather of eigenvectors. One 256-thread block per batch, ~131 KB LDS.
// ---------------------------------------------------------------------------
__global__ void k_jacobi(const float* __restrict__ Ag,
                         float* __restrict__ Yg,
                         float* __restrict__ LamG) {
  extern __shared__ float sm[];
  float* Am   = sm;                 // 16384 (A, later Ysorted staging)
  float* Vm   = sm + T2_;           // 16384
  float* carr = sm + 2 * T2_;       // 64
  float* sarr = carr + 64;          // 64
  float* keyv = sarr + 64;          // 128
  int*   perm = (int*)(keyv + 128); // 128
  int*   parr = perm + 128;         // 64
  int*   qarr = parr + 64;          // 64
  int*   sidx = qarr + 64;          // 128
  const int b = blockIdx.x, tid = threadIdx.x;
  g2l_async(Ag + (size_t)b * T2_, Am, T2_);
  for (int i = tid; i < T2_; i += 256)
    Vm[i] = ((i >> 7) == (i & 127)) ? 1.f : 0.f;
  if (tid < 128) perm[tid] = tid;
  copy_fence();

  for (int sweep = 0; sweep < 8; ++sweep) {
    for (int step = 0; step < 127; ++step) {
      if (tid < 64) {  // 64 disjoint rotations (tournament pairing)
        int p = perm[tid], q = perm[127 - tid];
        if (p > q) { int t = p; p = q; q = t; }
        const float app = Am[p * 128 + p];
        const float aqq = Am[q * 128 + q];
        const float apq = Am[p * 128 + q];
        float c = 1.f, s = 0.f;
        if (fabsf(apq) > 1e-30f) {
          const float tau = (aqq - app) / (2.f * apq);
          const float t2 = (tau >= 0.f ? 1.f : -1.f) /
                           (fabsf(tau) + sqrtf(1.f + tau * tau));
          c = rsqrtf(1.f + t2 * t2);
          s = t2 * c;
        }
        carr[tid] = c; sarr[tid] = s; parr[tid] = p; qarr[tid] = q;
      }
      __syncthreads();
      for (int i = tid; i < 8192; i += 256) {  // rows: B = J^T A
        const int r = i >> 7, k = i & 127;
        const int p = parr[r], q = qarr[r];
        const float c = carr[r], s = sarr[r];
        const float ap = Am[p * 128 + k], aq = Am[q * 128 + k];
        Am[p * 128 + k] = c * ap - s * aq;
        Am[q * 128 + k] = s * ap + c * aq;
      }
      __syncthreads();
      for (int i = tid; i < 8192; i += 256) {  // cols: A = B J, V = V J
        const int k = i & 127, r = i >> 7;
        const int p = parr[r], q = qarr[r];
        const float c = carr[r], s = sarr[r];
        const float ap = Am[k * 128 + p], aq = Am[k * 128 + q];
        Am[k * 128 + p] = c * ap - s * aq;
        Am[k * 128 + q] = s * ap + c * aq;
        const float vp = Vm[k * 128 + p], vq = Vm[k * 128 + q];
        Vm[k * 128 + p] = c * vp - s * vq;
        Vm[k * 128 + q] = s * vp + c * vq;
      }
      __syncthreads();
      const int oldv = (tid < 128) ? perm[tid] : 0;  // rotate schedule
      __syncthreads();
      if (tid >= 1 && tid < 128) perm[(tid == 127) ? 1 : tid + 1] = oldv;
      __syncthreads();
    }
  }

  // Ascending bitonic sort of eigenvalues, carrying column indices.
  if (tid < 128) { keyv[tid] = Am[tid * 128 + tid]; sidx[tid] = tid; }
  __syncthreads();
  for (int k = 2; k <= 128; k <<= 1) {
    for (int j = k >> 1; j > 0; j >>= 1) {
      if (tid < 128) {
        const int l = tid ^ j;
        if (l > tid) {
          const bool asc = ((tid & k) == 0);
          const float a0 = keyv[tid], a1 = keyv[l];
          if ((asc && a0 > a1) || (!asc && a0 < a1)) {
            keyv[tid] = a1; keyv[l] = a0;
            const int i0 = sidx[tid]; sidx[tid] = sidx[l]; sidx[l] = i0;
          }
        }
      }
      __syncthreads();
    }
  }
  if (tid < 128) LamG[b * 128 + tid] = fmaxf(keyv[tid], 1e-6f);
  __syncthreads();
  for (int i = tid; i < T2_; i += 256) {     // stage Ysorted into Am
    const int r = i >> 7, c = i & 127;
    Am[i] = Vm[r * 128 + sidx[c]];
  }
  __syncthreads();
  l2g_async(Yg + (size_t)b * T2_, Am, T2_);
  tdm_drain();
}

// ---------------------------------------------------------------------------
// Kernel 4: Phi = L^{-T} Ysorted; z = Phi^T (N x); z_mean, v, g. (192 KB LDS)
// ---------------------------------------------------------------------------
__global__ void k_phi_z(const float* __restrict__ LinvG,
                        const float* __restrict__ Yg,
                        const float* __restrict__ LamG,
                        const float* __restrict__ Ng,
                        const float* __restrict__ Xg,
                        float* __restrict__ Out) {
  extern __shared__ float sm[];
  float* b0 = sm;                 // Linv -> N -> g staging
  float* b1 = sm + T2_;           // Ys -> [x | Tmp] -> [z | Tmp]
  float* b2 = sm + 2 * T2_;       // Phi -> v staging
  float* zm = sm + 3 * T2_;       // 128
  float* red = zm + 128;          // 1
  const int b = blockIdx.x, tid = threadIdx.x;

  // Pull N and x toward L2/WGP$ while Linv/Ys DMA in and Phi computes.
  {
    const char* pn = (const char*)(Ng + (size_t)b * T2_);
    const char* px = (const char*)(Xg + (size_t)b * TD_);
    for (int i = tid; i < 512; i += 256) __builtin_prefetch(pn + i * 128, 0, 1);
    if (tid < 256) __builtin_prefetch(px + tid * 128, 0, 1);
  }
  g2l_async(LinvG + (size_t)b * T2_, b0, T2_);
  g2l_async(Yg + (size_t)b * T2_, b1, T2_);
  copy_fence();
  mm128_wmma<true, false>(b0, 128, b1, 128, b2, 128, 128);  // Phi = Linv^T*Ys
  __syncthreads();
  g2l_async(Ng + (size_t)b * T2_, b0, T2_);
  g2l_async(Xg + (size_t)b * TD_, b1, TD_);
  copy_fence();
  mm128_wmma<false, false>(b0, 128, b1, 64, b1 + TD_, 64, 64);  // Tmp = N*x
  __syncthreads();
  mm128_wmma<true, false>(b2, 128, b1 + TD_, 64, b1, 64, 64);   // z = Phi^T*Tmp
  __syncthreads();

  if (tid < 128) {
    float s = 0.f;
    for (int d = 0; d < 64; ++d) s += b1[tid * 64 + d];
    zm[tid] = s * (1.f / 64.f);
  }
  __syncthreads();
  if (tid == 0) {
    float s = 0.f;
    for (int i = 0; i < 128; ++i) s += zm[i] * zm[i];
    red[0] = s;
  }
  __syncthreads();
  const float znsq = red[0];

  // Stage v into b2 and g into b0, then DMA z/v/g out.
  for (int i = tid; i < TD_; i += 256) {
    const int r = i >> 6;
    b2[i] = (r == 0) ? 0.f : (b1[i] - b1[i - 64]);
  }
  const float* LamB = LamG + b * 128;
  for (int i = tid; i < T2_; i += 256) {
    const int r = i >> 7, c = i & 127;
    float val = 2.f * ALPHA * zm[r] * zm[c];
    if (r == c) val += LamB[r] + ALPHA * znsq;
    b0[i] = val;
  }
  __syncthreads();
  l2g_async(Out + (size_t)b * TD_, b1, TD_);                           // z
  l2g_async(Out + (size_t)B_ * TD_ + (size_t)b * TD_, b2, TD_);        // v
  l2g_async(Out + 2 * (size_t)B_ * TD_ + (size_t)b * T2_, b0, T2_);    // g
  tdm_drain();
}

// ---------------------------------------------------------------------------
// Kernel 5: Gamma[b,k,:,:] = (ALPHA/(Lam[b,k]+1e-6)) * I — pure streaming
// non-temporal B128 stores (268 MB > 192 MB L2 -> NT is the right hint;
// HBM-bound: ~11.5 us floor at 23.3 TB/s). One block per (b,k).
// ---------------------------------------------------------------------------
__global__ void k_gamma(const float* __restrict__ LamG, float* __restrict__ G) {
  const int blk = blockIdx.x;  // b*128 + k
  const float scale = ALPHA / (LamG[blk] + 1e-6f);
  v4f* out4 = (v4f*)(G + (size_t)blk * T2_);
  const int tid = threadIdx.x;
#pragma unroll
  for (int it = 0; it < 16; ++it) {
    const int i4 = it * 256 + tid;  // 0..4095
    const int row = i4 >> 5;        // 0..127
    const int j4 = i4 & 31;
    const float sc = ((row >> 2) == j4) ? scale : 0.f;
    const int r3 = row & 3;
    v4f v = {(r3 == 0) ? sc : 0.f, (r3 == 1) ? sc : 0.f,
             (r3 == 2) ? sc : 0.f, (r3 == 3) ? sc : 0.f};
    __builtin_nontemporal_store(v, out4 + i4);
  }
}

// ---------------------------------------------------------------------------
extern "C" void kernel_launch(void* const* d_in, const int* in_sizes, int n_in,
                              void* d_out, int out_size, void* d_ws, size_t ws_size,
                              hipStream_t stream) {
  (void)in_sizes; (void)n_in; (void)out_size; (void)ws_size;
  const float* x = (const float*)d_in[0];
  const float* K = (const float*)d_in[1];
  const float* N = (const float*)d_in[2];
  float* out = (float*)d_out;
  float* ws  = (float*)d_ws;

  // Workspace layout (floats): Linv | A | Ysorted | Lam  (~6.2 MB)
  float* Linv = ws;
  float* A    = ws + (size_t)B_ * T2_;
  float* Ys   = ws + 2 * (size_t)B_ * T2_;
  float* Lam  = ws + 3 * (size_t)B_ * T2_;

  k_chol_inv<<<B_, 128, 2 * T2_ * sizeof(float), stream>>>(N, Linv);
  k_form_A<<<B_, 256, 3 * T2_ * sizeof(float), stream>>>(Linv, K, A);
  k_jacobi<<<B_, 256, (2 * T2_ + 256 + 384 + 128) * sizeof(float), stream>>>(A, Ys, Lam);
  k_phi_z<<<B_, 256, (3 * T2_ + 160) * sizeof(float), stream>>>(Linv, Ys, Lam, N, x, out);
  k_gamma<<<B_ * T_, 256, 0, stream>>>(Lam, out + 4 * (size_t)B_ * TD_);
}